// mTANDClassifier_57312043598108
// MI455X (gfx1250) — compile-verified
//
#include <hip/hip_runtime.h>
#include <math.h>

#define B_ 16
#define T_ 200
#define TP_ 208        // T padded to multiple of 16 for WMMA M/N
#define F_ 41
#define ET_ 128
#define NH_ 128
#define NS_ 2
#define D_ 82          // 2*F
#define DP_ 96         // D padded to multiple of 16 for WMMA K

typedef __attribute__((ext_vector_type(2))) float v2f;
typedef __attribute__((ext_vector_type(8))) float v8f;

// ---------------------------------------------------------------------------
// te[t][e] for t in [0,TP): rows >= T are zero so padded GEMM rows are inert.
// ---------------------------------------------------------------------------
__global__ void k_time_embed(const float* __restrict__ W_lin, const float* __restrict__ b_lin,
                             const float* __restrict__ W_per, const float* __restrict__ b_per,
                             float* __restrict__ te) {
  int t = blockIdx.x;
  int e = threadIdx.x;
  if (t >= T_) { te[t * ET_ + e] = 0.f; return; }
  float tt = (float)t / (float)(T_ - 1);
  float v;
  if (e == 0) v = tt * W_lin[0] + b_lin[0];
  else        v = sinf(tt * W_per[e - 1] + b_per[e - 1]);
  te[t * ET_ + e] = v;
}

// Zero-padded copy of Wo (128 x 82) -> Wo_p (128 x 96)
__global__ void k_pad_wo(const float* __restrict__ Wo, float* __restrict__ Wo_p) {
  int n = blockIdx.x;        // 128
  int k = threadIdx.x;       // 96
  Wo_p[n * DP_ + k] = (k < D_) ? Wo[n * D_ + k] : 0.f;
}

// ---------------------------------------------------------------------------
// C[M,N] = alpha * (A[M,K] @ W[N,K]^T) + bias[N]
// Fully unguarded hot loop: M,N multiples of 16, K (template) multiple of 4,
// all operands padded in workspace.  One wave per 16x16 tile; per K-step:
// two global_load_b64 fragment loads + one V_WMMA_F32_16X16X4_F32.
// ---------------------------------------------------------------------------
template<int K>
__global__ void k_wmma_gemm_nt(const float* __restrict__ A, int lda,
                               const float* __restrict__ W, int ldw,
                               const float* __restrict__ bias,
                               float* __restrict__ C, int ldc,
                               int M, int N, float alpha) {
  int gw   = (blockIdx.x * blockDim.x + threadIdx.x) >> 5;
  int lane = threadIdx.x & 31;
  int mtiles = M >> 4;
  int ntiles = N >> 4;
  if (gw >= mtiles * ntiles) return;     // wave-uniform; EXEC all-1s below
  int mt = gw / ntiles;
  int nt = gw % ntiles;

  int l16 = lane & 15;
  int hi  = lane >> 4;                   // upper half-wave holds K+2,K+3
  const float* Arow = A + (size_t)(mt * 16 + l16) * lda + hi * 2;
  const float* Wrow = W + (size_t)(nt * 16 + l16) * ldw + hi * 2;

  v8f c = {};
#pragma unroll 8
  for (int k0 = 0; k0 < K; k0 += 4) {
    v2f a = *(const v2f*)(Arow + k0);
    v2f b = *(const v2f*)(Wrow + k0);
    c = __builtin_amdgcn_wmma_f32_16x16x4_f32(false, a, false, b, (short)0, c,
                                              false, false);
  }

  int n = nt * 16 + l16;
  float bn = (bias != nullptr) ? bias[n] : 0.f;
  float* Crow = C + (size_t)(mt * 16 + hi * 8) * ldc + n;
#pragma unroll
  for (int i = 0; i < 8; ++i)
    Crow[(size_t)i * ldc] = alpha * c[i] + bn;
}

// ---------------------------------------------------------------------------
// In-place LayerNorm over rows of width 128 (one block of 128 threads / row).
// ---------------------------------------------------------------------------
__global__ void k_ln_rows128(float* __restrict__ X, const float* __restrict__ g,
                             const float* __restrict__ beta) {
  __shared__ float red[128];
  int row = blockIdx.x;
  int tid = threadIdx.x;
  float x = X[(size_t)row * 128 + tid];
  red[tid] = x;
  __syncthreads();
  for (int s = 64; s > 0; s >>= 1) { if (tid < s) red[tid] += red[tid + s]; __syncthreads(); }
  float mean = red[0] * (1.f / 128.f);
  __syncthreads();
  float d = x - mean;
  red[tid] = d * d;
  __syncthreads();
  for (int s = 64; s > 0; s >>= 1) { if (tid < s) red[tid] += red[tid + s]; __syncthreads(); }
  float r = rsqrtf(red[0] * (1.f / 128.f) + 1e-5f);
  X[(size_t)row * 128 + tid] = d * r * g[tid] + beta[tid];
}

// ---------------------------------------------------------------------------
// Per-feature masked softmax attention + fused LayerNorm over D=82.
// Block = one (b, tq), thread = feature d.  Score row + mask[b] tile staged
// in LDS; online softmax over tk.  Output zero-padded to DP_=96 columns.
// ---------------------------------------------------------------------------
__global__ void k_attention(const float* __restrict__ x, const float* __restrict__ mask,
                            const float* __restrict__ scores,
                            const float* __restrict__ g_attn, const float* __restrict__ b_attn,
                            float* __restrict__ att_p) {
  __shared__ float sc[T_];
  __shared__ float smask[T_ * F_];       // 32.8 KB of the 320 KB WGP LDS
  __shared__ float red[128];
  int row = blockIdx.x;                  // b*T + tq
  int b   = row / T_;
  int tq  = row % T_;
  int tid = threadIdx.x;                 // 128 threads

  for (int i = tid; i < T_; i += 128) sc[i] = scores[(size_t)tq * TP_ + i];
  const float* mbase = mask + (size_t)b * T_ * F_;
  for (int i = tid; i < T_ * F_; i += 128) smask[i] = mbase[i];
  __syncthreads();

  float att = 0.f;
  if (tid < D_) {
    int f = (tid < F_) ? tid : tid - F_;
    const float* xrow = x + (size_t)b * T_ * F_ + f;
    float mx = -3.0e38f, ssum = 0.f, acc = 0.f;
    for (int tk = 0; tk < T_; ++tk) {
      float mval = smask[tk * F_ + f];
      float s = (mval == 0.f) ? -1e9f : sc[tk];
      float nmx = fmaxf(mx, s);
      float rs  = __expf(mx - nmx);      // rescale factor (0 on first iter)
      float e   = __expf(s - nmx);
      float v;
      if (tid < F_) { float xv = xrow[tk * F_]; v = (mval > 0.f) ? xv : 0.f; }
      else          { v = mval; }
      ssum = ssum * rs + e;
      acc  = acc  * rs + e * v;
      mx = nmx;
    }
    att = acc / ssum;
  }

  // fused LayerNorm over the 82 features
  red[tid] = (tid < D_) ? att : 0.f;
  __syncthreads();
  for (int s = 64; s > 0; s >>= 1) { if (tid < s) red[tid] += red[tid + s]; __syncthreads(); }
  float mean = red[0] * (1.f / (float)D_);
  __syncthreads();
  float d = att - mean;
  red[tid] = (tid < D_) ? d * d : 0.f;
  __syncthreads();
  for (int s = 64; s > 0; s >>= 1) { if (tid < s) red[tid] += red[tid + s]; __syncthreads(); }
  float r = rsqrtf(red[0] * (1.f / (float)D_) + 1e-5f);
  if (tid < DP_) {
    float outv = (tid < D_) ? (d * r * g_attn[tid] + b_attn[tid]) : 0.f;
    att_p[(size_t)row * DP_ + tid] = outv;
  }
}

// ---------------------------------------------------------------------------
// GRU scan over T (sequential) — single workgroup, state + gates in LDS.
// ---------------------------------------------------------------------------
__global__ void k_gru(const float* __restrict__ h, const float* __restrict__ W_ih,
                      const float* __restrict__ W_hh, const float* __restrict__ b_ih,
                      const float* __restrict__ b_hh, float* __restrict__ hT) {
  __shared__ float hp[B_ * NH_];         // 8 KB
  __shared__ float gx[B_ * 3 * NH_];     // 24 KB
  __shared__ float gh[B_ * 3 * NH_];     // 24 KB
  int tid = threadIdx.x;                 // 512 threads
  for (int i = tid; i < B_ * NH_; i += 512) hp[i] = 0.f;

  for (int t = 0; t < T_; ++t) {
    __syncthreads();                     // hp ready / visible
    for (int idx = tid; idx < B_ * 3 * NH_; idx += 512) {
      int b = idx / (3 * NH_);
      int j = idx % (3 * NH_);
      const float* hin = h + ((size_t)b * T_ + t) * NH_;
      const float* wi  = W_ih + (size_t)j * NH_;
      const float* wh  = W_hh + (size_t)j * NH_;
      const float* hpb = hp + b * NH_;
      float sx = b_ih[j], sh = b_hh[j];
      for (int k = 0; k < NH_; ++k) {
        sx = fmaf(hin[k], wi[k], sx);
        sh = fmaf(hpb[k], wh[k], sh);
      }
      gx[idx] = sx; gh[idx] = sh;
    }
    __syncthreads();                     // gates ready
    for (int idx = tid; idx < B_ * NH_; idx += 512) {
      int b = idx >> 7;
      int j = idx & 127;
      const float* gxb = gx + b * 3 * NH_;
      const float* ghb = gh + b * 3 * NH_;
      float r = 1.f / (1.f + __expf(-(gxb[j]           + ghb[j])));
      float z = 1.f / (1.f + __expf(-(gxb[NH_ + j]     + ghb[NH_ + j])));
      float n = tanhf(gxb[2 * NH_ + j] + r * ghb[2 * NH_ + j]);
      hp[idx] = (1.f - z) * n + z * hp[idx];   // only owner thread touches idx
    }
  }
  __syncthreads();
  for (int i = tid; i < B_ * NH_; i += 512) hT[i] = hp[i];
}

// ---------------------------------------------------------------------------
// Classifier MLP: relu(relu(hT W1ᵀ+b1) W2ᵀ+b2) W3ᵀ+b3  -> (16,2)
// ---------------------------------------------------------------------------
__global__ void k_classifier(const float* __restrict__ hT,
                             const float* __restrict__ Wc1, const float* __restrict__ bc1,
                             const float* __restrict__ Wc2, const float* __restrict__ bc2,
                             const float* __restrict__ Wc3, const float* __restrict__ bc3,
                             float* __restrict__ out) {
  __shared__ float s0[B_ * NH_];
  __shared__ float s1[B_ * NH_];
  int tid = threadIdx.x;                 // 256
  for (int i = tid; i < B_ * NH_; i += 256) s0[i] = hT[i];
  __syncthreads();
  for (int idx = tid; idx < B_ * NH_; idx += 256) {
    int b = idx >> 7, j = idx & 127;
    float s = bc1[j];
    for (int k = 0; k < NH_; ++k) s = fmaf(s0[b * NH_ + k], Wc1[j * NH_ + k], s);
    s1[idx] = fmaxf(s, 0.f);
  }
  __syncthreads();
  for (int idx = tid; idx < B_ * NH_; idx += 256) {
    int b = idx >> 7, j = idx & 127;
    float s = bc2[j];
    for (int k = 0; k < NH_; ++k) s = fmaf(s1[b * NH_ + k], Wc2[j * NH_ + k], s);
    s0[idx] = fmaxf(s, 0.f);
  }
  __syncthreads();
  for (int idx = tid; idx < B_ * NS_; idx += 256) {
    int b = idx / NS_, j = idx % NS_;
    float s = bc3[j];
    for (int k = 0; k < NH_; ++k) s = fmaf(s0[b * NH_ + k], Wc3[j * NH_ + k], s);
    out[idx] = s;
  }
}

// ---------------------------------------------------------------------------
extern "C" void kernel_launch(void* const* d_in, const int* in_sizes, int n_in,
                              void* d_out, int out_size, void* d_ws, size_t ws_size,
                              hipStream_t stream) {
  const float* x     = (const float*)d_in[0];
  const float* mask  = (const float*)d_in[1];
  const float* W_lin = (const float*)d_in[2];
  const float* b_lin = (const float*)d_in[3];
  const float* W_per = (const float*)d_in[4];
  const float* b_per = (const float*)d_in[5];
  const float* Wq    = (const float*)d_in[6];
  const float* bq    = (const float*)d_in[7];
  const float* gq    = (const float*)d_in[8];
  const float* betaq = (const float*)d_in[9];
  const float* Wk    = (const float*)d_in[10];
  const float* bk    = (const float*)d_in[11];
  const float* gk    = (const float*)d_in[12];
  const float* betak = (const float*)d_in[13];
  const float* g_at  = (const float*)d_in[14];
  const float* b_at  = (const float*)d_in[15];
  const float* Wo    = (const float*)d_in[16];
  const float* bo    = (const float*)d_in[17];
  const float* go    = (const float*)d_in[18];
  const float* betao = (const float*)d_in[19];
  const float* W_ih  = (const float*)d_in[20];
  const float* W_hh  = (const float*)d_in[21];
  const float* b_ih  = (const float*)d_in[22];
  const float* b_hh  = (const float*)d_in[23];
  const float* Wc1   = (const float*)d_in[24];
  const float* bc1   = (const float*)d_in[25];
  const float* Wc2   = (const float*)d_in[26];
  const float* bc2   = (const float*)d_in[27];
  const float* Wc3   = (const float*)d_in[28];
  const float* bc3   = (const float*)d_in[29];
  float* out = (float*)d_out;

  float* ws     = (float*)d_ws;
  float* te     = ws;  ws += TP_ * ET_;       // 26624
  float* qb     = ws;  ws += TP_ * ET_;       // 26624
  float* kb     = ws;  ws += TP_ * ET_;       // 26624
  float* scores = ws;  ws += TP_ * TP_;       // 43264
  float* wo_p   = ws;  ws += NH_ * DP_;       // 12288
  float* att_p  = ws;  ws += B_ * T_ * DP_;   // 307200
  float* hbuf   = ws;  ws += B_ * T_ * NH_;   // 409600
  float* hT     = ws;  ws += B_ * NH_;        // 2048   (total ~3.45 MB)

  k_time_embed<<<TP_, ET_, 0, stream>>>(W_lin, b_lin, W_per, b_per, te);
  k_pad_wo<<<NH_, DP_, 0, stream>>>(Wo, wo_p);

  auto grid_for = [](int M, int N) {
    int waves = (M / 16) * (N / 16);
    return (waves * 32 + 255) / 256;
  };

  // q/k projections: (208,128) @ (128,128)^T  (pad rows are zero -> bias)
  k_wmma_gemm_nt<ET_><<<grid_for(TP_, ET_), 256, 0, stream>>>(
      te, ET_, Wq, ET_, bq, qb, ET_, TP_, ET_, 1.f);
  k_ln_rows128<<<T_, 128, 0, stream>>>(qb, gq, betaq);
  k_wmma_gemm_nt<ET_><<<grid_for(TP_, ET_), 256, 0, stream>>>(
      te, ET_, Wk, ET_, bk, kb, ET_, TP_, ET_, 1.f);
  k_ln_rows128<<<T_, 128, 0, stream>>>(kb, gk, betak);

  // scores = q @ k^T / sqrt(128): (208,208); pad band never read downstream
  k_wmma_gemm_nt<ET_><<<grid_for(TP_, TP_), 256, 0, stream>>>(
      qb, ET_, kb, ET_, nullptr, scores, TP_, TP_, TP_, 0.0883883476483184f);

  // masked per-feature softmax attention + LN  -> att_p (B*T, 96)
  k_attention<<<B_ * T_, 128, 0, stream>>>(x, mask, scores, g_at, b_at, att_p);

  // output projection + LN: (3200,96) @ (128,96)^T -> (3200,128)
  k_wmma_gemm_nt<DP_><<<grid_for(B_ * T_, NH_), 256, 0, stream>>>(
      att_p, DP_, wo_p, DP_, bo, hbuf, NH_, B_ * T_, NH_, 1.f);
  k_ln_rows128<<<B_ * T_, 128, 0, stream>>>(hbuf, go, betao);

  // sequential GRU over T
  k_gru<<<1, 512, 0, stream>>>(hbuf, W_ih, W_hh, b_ih, b_hh, hT);

  // classifier head -> (16,2)
  k_classifier<<<1, 256, 0, stream>>>(hT, Wc1, bc1, Wc2, bc2, Wc3, bc3, out);
}